// GATLayer_42417097015743
// MI455X (gfx1250) — compile-verified
//
#include <hip/hip_runtime.h>
#include <math.h>

typedef __attribute__((ext_vector_type(2))) float v2f;
typedef __attribute__((ext_vector_type(8))) float v8f;

#define NEG_SLOPE 0.2f

__device__ __forceinline__ float lrelu(float s) { return s > 0.f ? s : NEG_SLOPE * s; }

// Order-preserving float -> uint encoding for atomic max on floats of any sign.
__device__ __forceinline__ unsigned fenc(float f) {
  unsigned u = __float_as_uint(f);
  return (u & 0x80000000u) ? ~u : (u | 0x80000000u);
}
__device__ __forceinline__ float fdec(unsigned k) {
  unsigned u = (k & 0x80000000u) ? (k ^ 0x80000000u) : ~k;
  return __uint_as_float(u);
}

// ---------------------------------------------------------------------------
// Init: out = bias (broadcast), amax keys = 0 (== -inf), denom = 0
// ---------------------------------------------------------------------------
__global__ void gat_init(float* __restrict__ out, const float* __restrict__ bias,
                         unsigned* __restrict__ amax, float* __restrict__ denom, int n) {
  int i = blockIdx.x * blockDim.x + threadIdx.x;
  if (i < n * 128) out[i] = bias[i & 127];
  if (i < n * 8) { amax[i] = 0u; denom[i] = 0.f; }
}

// ---------------------------------------------------------------------------
// GEMM h = x @ W with V_WMMA_F32_16X16X4_F32, fused attention-logit epilogue.
// Block = 256 threads = 8 waves; wave w owns head w (cols [16w,16w+16)).
// Each block processes 4 row-tiles of 16 rows. B fragments live in registers
// for the whole kernel; the 16x128 x-tile is staged in LDS per tile.
// ---------------------------------------------------------------------------
__global__ __launch_bounds__(256)
void gat_gemm(const float* __restrict__ x, const float* __restrict__ W,
              const float* __restrict__ attS, const float* __restrict__ attD,
              float* __restrict__ h, float* __restrict__ aS, float* __restrict__ aD,
              int ntiles) {
  __shared__ float lX[16 * 128];   // 8 KB x-tile
  const int tid  = threadIdx.x;
  const int lane = tid & 31;
  const int wave = tid >> 5;            // == head (0..7)
  const int m    = lane & 15;           // A: row within tile / B,C: column within head
  const int kh   = (lane >> 4) << 1;    // K sub-offset per 32-bit WMMA layout
  const int col0 = wave * 16;

  const float attSv = attS[col0 + m];
  const float attDv = attD[col0 + m];

  // Preload B fragments: lane holds B[k][n] for k = 4q+kh+{0,1}, n = m. 64 VGPRs.
  v2f bfrag[32];
#pragma unroll
  for (int q = 0; q < 32; ++q) {
    int k = 4 * q + kh;
    bfrag[q][0] = W[(size_t)k * 128 + col0 + m];
    bfrag[q][1] = W[(size_t)(k + 1) * 128 + col0 + m];
  }

#pragma unroll 1
  for (int ti = 0; ti < 4; ++ti) {
    int t = blockIdx.x * 4 + ti;
    if (t >= ntiles) break;                 // uniform across block
    size_t row0 = (size_t)t * 16;

    __syncthreads();                        // WAR guard on lX
    for (int i = tid * 4; i < 16 * 128; i += 256 * 4)
      *(float4*)(lX + i) = *(const float4*)(x + row0 * 128 + i);
    __syncthreads();

    v8f c = {};
#pragma unroll
    for (int q = 0; q < 32; ++q) {
      v2f a = *(const v2f*)(lX + m * 128 + 4 * q + kh);   // ds_load_b64
      c = __builtin_amdgcn_wmma_f32_16x16x4_f32(false, a, false, bfrag[q],
                                                (short)0, c, false, false);
    }

    // Epilogue: store h tile; reduce c[r]*att over the 16 columns of this head
    // to get per-row logits. C layout: vgpr r, lanes0-15 -> M=r, lanes16-31 -> M=8+r.
    const int rbase = (lane >> 4) * 8;
#pragma unroll
    for (int r = 0; r < 8; ++r) {
      size_t row = row0 + rbase + r;
      h[row * 128 + col0 + m] = c[r];
      float vs = c[r] * attSv;
      float vd = c[r] * attDv;
      vs += __shfl_xor(vs, 1, 32); vs += __shfl_xor(vs, 2, 32);
      vs += __shfl_xor(vs, 4, 32); vs += __shfl_xor(vs, 8, 32);
      vd += __shfl_xor(vd, 1, 32); vd += __shfl_xor(vd, 2, 32);
      vd += __shfl_xor(vd, 4, 32); vd += __shfl_xor(vd, 8, 32);
      if (m == 0) {
        aS[row * 8 + wave] = vs;
        aD[row * 8 + wave] = vd;
      }
    }
  }
}

// ---------------------------------------------------------------------------
// Pass 1: segment max of leaky_relu(a_src[src]+a_dst[dst]) per (dst, head)
// ---------------------------------------------------------------------------
__global__ void gat_edge_max(const int* __restrict__ ei, const float* __restrict__ aS,
                             const float* __restrict__ aD, unsigned* __restrict__ amax,
                             int E, int total) {
  int e = blockIdx.x * blockDim.x + threadIdx.x;
  if (e >= total) return;
  int src = (e < E) ? ei[e]     : (e - E);
  int dst = (e < E) ? ei[E + e] : (e - E);
  const float4* ps = (const float4*)(aS + (size_t)src * 8);
  const float4* pd = (const float4*)(aD + (size_t)dst * 8);
  float4 s0 = ps[0], s1 = ps[1], d0 = pd[0], d1 = pd[1];
  float v[8] = { s0.x + d0.x, s0.y + d0.y, s0.z + d0.z, s0.w + d0.w,
                 s1.x + d1.x, s1.y + d1.y, s1.z + d1.z, s1.w + d1.w };
  unsigned* base = amax + (size_t)dst * 8;
#pragma unroll
  for (int hh = 0; hh < 8; ++hh) atomicMax(base + hh, fenc(lrelu(v[hh])));
}

// ---------------------------------------------------------------------------
// Pass 2: denom[dst,h] += exp(alpha - amax[dst,h])
// ---------------------------------------------------------------------------
__global__ void gat_edge_sum(const int* __restrict__ ei, const float* __restrict__ aS,
                             const float* __restrict__ aD, const unsigned* __restrict__ amax,
                             float* __restrict__ denom, int E, int total) {
  int e = blockIdx.x * blockDim.x + threadIdx.x;
  if (e >= total) return;
  int src = (e < E) ? ei[e]     : (e - E);
  int dst = (e < E) ? ei[E + e] : (e - E);
  const float4* ps = (const float4*)(aS + (size_t)src * 8);
  const float4* pd = (const float4*)(aD + (size_t)dst * 8);
  float4 s0 = ps[0], s1 = ps[1], d0 = pd[0], d1 = pd[1];
  float v[8] = { s0.x + d0.x, s0.y + d0.y, s0.z + d0.z, s0.w + d0.w,
                 s1.x + d1.x, s1.y + d1.y, s1.z + d1.z, s1.w + d1.w };
  const unsigned* mk = amax + (size_t)dst * 8;
  float* dn = denom + (size_t)dst * 8;
#pragma unroll
  for (int hh = 0; hh < 8; ++hh) {
    float ex = __expf(lrelu(v[hh]) - fdec(mk[hh]));
    atomicAdd(dn + hh, ex);
  }
}

// ---------------------------------------------------------------------------
// Pass 3: out[dst] += h[src] * softmax_weight. One wave per edge; each lane
// owns 4 channels (float4), head = lane>>2.
// ---------------------------------------------------------------------------
__global__ void gat_aggregate(const int* __restrict__ ei, const float* __restrict__ h,
                              const float* __restrict__ aS, const float* __restrict__ aD,
                              const unsigned* __restrict__ amax, const float* __restrict__ denom,
                              float* __restrict__ out, int E, int total) {
  long long gid = (long long)blockIdx.x * blockDim.x + threadIdx.x;
  int e    = (int)(gid >> 5);
  int lane = (int)(gid & 31);
  if (e >= total) return;
  int src = (e < E) ? ei[e]     : (e - E);
  int dst = (e < E) ? ei[E + e] : (e - E);
  int head = lane >> 2;
  float s  = lrelu(aS[(size_t)src * 8 + head] + aD[(size_t)dst * 8 + head]);
  float ex = __expf(s - fdec(amax[(size_t)dst * 8 + head]));
  float wgt = ex / (denom[(size_t)dst * 8 + head] + 1e-16f);
  float4 hv = *(const float4*)(h + (size_t)src * 128 + lane * 4);
  float* ob = out + (size_t)dst * 128 + lane * 4;
  atomicAdd(ob + 0, hv.x * wgt);
  atomicAdd(ob + 1, hv.y * wgt);
  atomicAdd(ob + 2, hv.z * wgt);
  atomicAdd(ob + 3, hv.w * wgt);
}

// ---------------------------------------------------------------------------
extern "C" void kernel_launch(void* const* d_in, const int* in_sizes, int n_in,
                              void* d_out, int out_size, void* d_ws, size_t ws_size,
                              hipStream_t stream) {
  const float* x    = (const float*)d_in[0];   // [N,128]
  const int*   ei   = (const int*)  d_in[1];   // [2,E]
  const float* W    = (const float*)d_in[2];   // [128,128]
  const float* attS = (const float*)d_in[3];   // [8,16]
  const float* attD = (const float*)d_in[4];   // [8,16]
  const float* bias = (const float*)d_in[5];   // [128]

  const int N = in_sizes[0] / 128;
  const int E = in_sizes[1] / 2;
  const int total = E + N;                     // edges + self loops

  float* out = (float*)d_out;
  char* ws = (char*)d_ws;
  float*    h     = (float*)   (ws);                        // N*128 f32 = 51.2 MB
  float*    aS    = (float*)   (ws + (size_t)51200000);     // N*8
  float*    aD    = (float*)   (ws + (size_t)54400000);     // N*8
  unsigned* amax  = (unsigned*)(ws + (size_t)57600000);     // N*8
  float*    denom = (float*)   (ws + (size_t)60800000);     // N*8

  // 1) init out=bias, amax=-inf keys, denom=0
  gat_init<<<(N * 128 + 255) / 256, 256, 0, stream>>>(out, bias, amax, denom, N);

  // 2) h = x @ W (f32 WMMA) with fused a_src/a_dst epilogue
  int ntiles = (N + 15) / 16;
  gat_gemm<<<(ntiles + 3) / 4, 256, 0, stream>>>(x, W, attS, attD, h, aS, aD, ntiles);

  // 3) segment max
  gat_edge_max<<<(total + 255) / 256, 256, 0, stream>>>(ei, aS, aD, amax, E, total);

  // 4) segment sum of exp
  gat_edge_sum<<<(total + 255) / 256, 256, 0, stream>>>(ei, aS, aD, amax, denom, E, total);

  // 5) weighted aggregation, one wave per edge
  long long aggThreads = (long long)total * 32;
  int aggBlocks = (int)((aggThreads + 255) / 256);
  gat_aggregate<<<aggBlocks, 256, 0, stream>>>(ei, h, aS, aD, amax, denom, out, E, total);
}